// SynapseNetwork_42494406426725
// MI455X (gfx1250) — compile-verified
//
#include <hip/hip_runtime.h>
#include <hip/hip_bf16.h>
#include <stdint.h>

typedef float f4  __attribute__((ext_vector_type(4)));
typedef float v2f __attribute__((ext_vector_type(2)));
typedef float v8f __attribute__((ext_vector_type(8)));

#define K1 0.05f                    // DT/TAU
#define K2 1.6666666666666666e-4f   // DT/TAU_D
#define ROWS_PER_TILE 64

// ---------------------------------------------------------------------------
// Streaming kernel: bandwidth-bound elementwise update + per-column partials.
// Each thread: 4 consecutive columns (float4), 64-row tile. Non-temporal
// 128-bit loads/stores for the 1.1GB one-shot stream; the small partials
// intermediate is stored temporally so it stays resident in the 192MB L2
// for the reduction pass.
// ---------------------------------------------------------------------------
template <bool USE_ATOMIC>
__global__ __launch_bounds__(256) void syn_main(
    const float* __restrict__ spike, const float* __restrict__ s,
    const float* __restrict__ D, const unsigned char* __restrict__ syns,
    float* __restrict__ s_new, float* __restrict__ D_new,
    float* __restrict__ f_sink,   // partials [tile][col] (or f_all if atomic)
    int rows, int cols)
{
    const int col = (blockIdx.x * blockDim.x + threadIdx.x) * 4;
    if (col >= cols) return;
    const int tile   = blockIdx.y;
    const int row0   = tile * ROWS_PER_TILE;
    const int rowEnd = min(row0 + ROWS_PER_TILE, rows);

    f4 acc = {0.f, 0.f, 0.f, 0.f};

#pragma unroll 4
    for (int row = row0; row < rowEnd; ++row) {
        const size_t idx = (size_t)row * (size_t)cols + (size_t)col;
        const float  spk = spike[row];   // uniform -> scalar load (K$)
        f4 sv = __builtin_nontemporal_load((const f4*)(s + idx));
        f4 Dv = __builtin_nontemporal_load((const f4*)(D + idx));
        unsigned int mraw =
            __builtin_nontemporal_load((const unsigned int*)(syns + idx));
        f4 sn, Dn;
#pragma unroll
        for (int i = 0; i < 4; ++i) {
            float mi  = (float)((mraw >> (8 * i)) & 0xFFu);  // 0.0 or 1.0
            float si  = sv[i];
            float Di  = Dv[i];
            float ds  = __builtin_fmaf(si, -K1, spk);   // spike - s*DT/TAU
            float sni = __builtin_fmaf(mi, ds, si);     // s + m*ds
            float dD  = __builtin_fmaf(Di, -K2, K2);    // (1-D)*DT/TAU_D
            float Dni = __builtin_fmaf(mi, dD, Di);     // D + m*dD
            sn[i] = sni;
            Dn[i] = Dni;
            acc[i] = __builtin_fmaf(mi, Dni * sni, acc[i]);
        }
        __builtin_nontemporal_store(sn, (f4*)(s_new + idx));
        __builtin_nontemporal_store(Dn, (f4*)(D_new + idx));
    }

    if (USE_ATOMIC) {
        atomicAdd(&f_sink[col + 0], acc[0]);
        atomicAdd(&f_sink[col + 1], acc[1]);
        atomicAdd(&f_sink[col + 2], acc[2]);
        atomicAdd(&f_sink[col + 3], acc[3]);
    } else {
        // Temporal store: keep the 4MB partials in L2 for the reduce pass.
        *(f4*)(f_sink + (size_t)tile * (size_t)cols + (size_t)col) = acc;
    }
}

// ---------------------------------------------------------------------------
// WMMA column-sum: f_all = ones(1 x numTiles) @ partials(numTiles x cols).
// A = all-ones 16x4 -> D[m,n] = sum_k B[k,n]; any K permutation in the HW
// A/B layouts is irrelevant, only N = lane&15 matters (honored: both lane
// halves load the same 16-column group). One wave -> 16 columns, 4 tile-rows
// folded per v_wmma_f32_16x16x4_f32. Hot loop uses pointer increments (no
// 64-bit muls, no clamps); ragged tail handled by one masked iteration.
// Result row M=0 lives in accumulator VGPR0 of lanes 0..15.
// ---------------------------------------------------------------------------
__global__ __launch_bounds__(256) void syn_reduce(
    const float* __restrict__ partials, float* __restrict__ f_all,
    int numTiles, int cols)
{
    const int gtid = blockIdx.x * blockDim.x + threadIdx.x;
    const int wave = gtid >> 5;
    const int lane = gtid & 31;
    const int half = lane >> 4;          // which 16-lane half
    const int col  = wave * 16 + (lane & 15);
    const int colIdx = (col < cols) ? col : (cols - 1);  // clamp, no EXEC div

    v8f c = {0.f, 0.f, 0.f, 0.f, 0.f, 0.f, 0.f, 0.f};
    v2f a = {1.0f, 1.0f};                // all-ones A matrix

    // rows visited per iteration: k + 2*half and k + 2*half + 1
    const float* p    = partials + (size_t)(2 * half) * (size_t)cols + colIdx;
    const size_t step = 4 * (size_t)cols;
    const int  kFull  = numTiles & ~3;

    for (int k = 0; k < kFull; k += 4) {
        v2f b;
        b.x = p[0];
        b.y = p[cols];
        // D = A(16x4 ones) * B(4x16 partials) + C  -> column sums
        c = __builtin_amdgcn_wmma_f32_16x16x4_f32(false, a, false, b,
                                                  (short)0, c, false, false);
        p += step;
    }
    if (numTiles & 3) {                  // ragged tail (masked values, clamped
        const int r0  = kFull + 2 * half; // addresses -> EXEC stays all-ones)
        const int r1  = r0 + 1;
        const int r0c = min(r0, numTiles - 1);
        const int r1c = min(r1, numTiles - 1);
        const float* base = partials + colIdx;
        v2f b;
        b.x = base[(size_t)r0c * (size_t)cols];
        b.y = base[(size_t)r1c * (size_t)cols];
        b.x = (r0 < numTiles) ? b.x : 0.0f;   // v_cndmask, EXEC untouched
        b.y = (r1 < numTiles) ? b.y : 0.0f;
        c = __builtin_amdgcn_wmma_f32_16x16x4_f32(false, a, false, b,
                                                  (short)0, c, false, false);
    }
    if (lane < 16 && col < cols) f_all[col] = c[0];
}

__global__ void zero_f(float* __restrict__ p, int n) {
    int i = blockIdx.x * blockDim.x + threadIdx.x;
    if (i < n) p[i] = 0.0f;
}

extern "C" void kernel_launch(void* const* d_in, const int* in_sizes, int n_in,
                              void* d_out, int out_size, void* d_ws,
                              size_t ws_size, hipStream_t stream) {
    const float*         spike = (const float*)d_in[0];
    const float*         s     = (const float*)d_in[1];
    const float*         D     = (const float*)d_in[2];
    const unsigned char* syns  = (const unsigned char*)d_in[3];  // jnp bool = 1B

    const int rows = in_sizes[0];
    const int cols = (int)((long long)in_sizes[1] / rows);

    float* out   = (float*)d_out;
    float* s_new = out;
    float* D_new = out + (size_t)rows * (size_t)cols;
    float* f_all = out + 2 * (size_t)rows * (size_t)cols;

    const int    numTiles = (rows + ROWS_PER_TILE - 1) / ROWS_PER_TILE;
    const size_t needed   = (size_t)numTiles * (size_t)cols * sizeof(float);

    dim3 block(256);
    const int colBlocks = (cols + 256 * 4 - 1) / (256 * 4);
    dim3 grid(colBlocks, numTiles);

    if (ws_size >= needed) {
        float* partials = (float*)d_ws;
        syn_main<false><<<grid, block, 0, stream>>>(spike, s, D, syns, s_new,
                                                    D_new, partials, rows, cols);
        const int waves   = (cols + 15) / 16;
        const int threads = waves * 32;
        const int rblocks = (threads + 255) / 256;
        syn_reduce<<<rblocks, 256, 0, stream>>>(partials, f_all, numTiles, cols);
    } else {
        zero_f<<<(cols + 255) / 256, 256, 0, stream>>>(f_all, cols);
        syn_main<true><<<grid, block, 0, stream>>>(spike, s, D, syns, s_new,
                                                   D_new, f_all, rows, cols);
    }
}